// MedianBlur_37434934952174
// MI455X (gfx1250) — compile-verified
//
#include <hip/hip_runtime.h>
#include <hip/hip_bf16.h>
#include <stdint.h>

// 3x3 median blur, fp32, zero padding. Input/output: [16, 3, 512, 512].
// Strategy: async global->LDS tile staging (CDNA5 ASYNCcnt path), then a
// shared-column median-of-9 network built from v_min3/v_max3/v_med3
// (~8.5 VALU ops/output), float4 I/O.

#define H 512
#define W 512
#define TILE_W 128   // output tile width  (floats)
#define TILE_H 16    // output tile height (rows)
#define LDS_W 136    // padded row stride in floats (16B aligned, > 133)
#define LDS_ROWS 18  // TILE_H + 2 halo rows

__device__ __forceinline__ void async_ld_b128(uint32_t lds_off, const float* g) {
  asm volatile("global_load_async_to_lds_b128 %0, %1, off"
               :: "v"(lds_off), "v"(g) : "memory");
}
__device__ __forceinline__ void async_ld_b32(uint32_t lds_off, const float* g) {
  asm volatile("global_load_async_to_lds_b32 %0, %1, off"
               :: "v"(lds_off), "v"(g) : "memory");
}
__device__ __forceinline__ void wait_async0() {
  asm volatile("s_wait_asynccnt 0" ::: "memory");
}

__device__ __forceinline__ uint32_t lds_off_of(const float* p) {
  // Generic pointers to LDS carry the LDS byte offset in the low 32 bits.
  return (uint32_t)(uintptr_t)p;
}

// Single-instruction 3-input min/max/med (ISel wasn't folding the chains).
__device__ __forceinline__ float min3f(float a, float b, float c) {
  float d;
  asm("v_min3_num_f32 %0, %1, %2, %3" : "=v"(d) : "v"(a), "v"(b), "v"(c));
  return d;
}
__device__ __forceinline__ float max3f(float a, float b, float c) {
  float d;
  asm("v_max3_num_f32 %0, %1, %2, %3" : "=v"(d) : "v"(a), "v"(b), "v"(c));
  return d;
}
__device__ __forceinline__ float med3f(float a, float b, float c) {
  return __builtin_amdgcn_fmed3f(a, b, c);
}

__device__ __forceinline__ void load_row6(const float* p, float* w) {
  // p points at LDS column (4*tx + 3); p+1 is 16B aligned.
  w[0] = p[0];
  float4 q = *(const float4*)(p + 1);
  w[1] = q.x; w[2] = q.y; w[3] = q.z; w[4] = q.w;
  w[5] = p[5];
}

__global__ void __launch_bounds__(256)
median3x3_kernel(const float* __restrict__ x, float* __restrict__ out) {
  __shared__ __align__(16) float tile[LDS_ROWS * LDS_W];

  const int tid = threadIdx.x;
  const int x0 = blockIdx.x * TILE_W;
  const int y0 = blockIdx.y * TILE_H;
  const size_t plane_base = (size_t)blockIdx.z * (size_t)(H * W);
  const float* plane = x + plane_base;
  float* outp = out + plane_base;

  // ---- zero-fill LDS halo cells that fall outside the image --------------
  // (disjoint from the addresses written by the async loads below)
  if (y0 == 0) {
    for (int c = tid; c < 130; c += 256) tile[0 * LDS_W + 3 + c] = 0.0f;
  }
  if (y0 + TILE_H == H) {
    for (int c = tid; c < 130; c += 256) tile[(LDS_ROWS - 1) * LDS_W + 3 + c] = 0.0f;
  }
  if (x0 == 0 && tid < LDS_ROWS) tile[tid * LDS_W + 3] = 0.0f;
  if (x0 + TILE_W == W && tid < LDS_ROWS) tile[tid * LDS_W + 132] = 0.0f;

  // ---- async global -> LDS: interior (18 rows x 32 float4) ----------------
  // LDS column mapping: global col gx -> LDS col (gx - x0 + 4); interior
  // starts at col 4 (16B aligned), halos at cols 3 and 132.
  for (int idx = tid; idx < LDS_ROWS * 32; idx += 256) {
    const int r = idx >> 5;       // LDS row 0..17
    const int t = idx & 31;       // float4 index within row
    const int gy = y0 - 1 + r;
    if (gy >= 0 && gy < H) {
      const float* g = plane + (size_t)gy * W + (x0 + 4 * t);
      async_ld_b128(lds_off_of(&tile[r * LDS_W + 4 + 4 * t]), g);
    }
  }
  // halo columns (one float each side per row)
  if (tid < 2 * LDS_ROWS) {
    const int r = tid >> 1;
    const int side = tid & 1;
    const int gy = y0 - 1 + r;
    const int gx = side ? (x0 + TILE_W) : (x0 - 1);
    if (gy >= 0 && gy < H && gx >= 0 && gx < W) {
      const int c = side ? 132 : 3;
      async_ld_b32(lds_off_of(&tile[r * LDS_W + c]), plane + (size_t)gy * W + gx);
    }
  }

  wait_async0();     // ASYNCcnt == 0: this wave's LDS writes have landed
  __syncthreads();   // all waves' writes visible

  // ---- compute: each thread -> 2 row-groups x 4 outputs -------------------
  const int tx = tid & 31;  // 0..31 -> 4 consecutive output columns
  const int ty = tid >> 5;  // 0..7
  const int col = 4 * tx;

#pragma unroll
  for (int p = 0; p < 2; ++p) {
    const int ry = ty + 8 * p;  // output row within tile (0..15)
    const float* r0 = &tile[(ry + 0) * LDS_W + col + 3];
    const float* r1 = &tile[(ry + 1) * LDS_W + col + 3];
    const float* r2 = &tile[(ry + 2) * LDS_W + col + 3];

    float w0[6], w1[6], w2[6];
    load_row6(r0, w0);
    load_row6(r1, w1);
    load_row6(r2, w2);

    // Per-column vertical sort3 via 3-input ops (3 instrs/column).
    float lo[6], mi[6], hi[6];
#pragma unroll
    for (int c = 0; c < 6; ++c) {
      lo[c] = min3f(w0[c], w1[c], w2[c]);
      mi[c] = med3f(w0[c], w1[c], w2[c]);
      hi[c] = max3f(w0[c], w1[c], w2[c]);
    }

    // median9 = med3(max3(los), med3(mids), min3(his))  (4 instrs/output).
    float o[4];
#pragma unroll
    for (int k = 0; k < 4; ++k) {
      const float L  = max3f(lo[k], lo[k + 1], lo[k + 2]);
      const float Hi = min3f(hi[k], hi[k + 1], hi[k + 2]);
      const float M  = med3f(mi[k], mi[k + 1], mi[k + 2]);
      o[k] = med3f(L, M, Hi);
    }

    float4 ov = make_float4(o[0], o[1], o[2], o[3]);
    *(float4*)&outp[(size_t)(y0 + ry) * W + x0 + col] = ov;
  }
}

extern "C" void kernel_launch(void* const* d_in, const int* in_sizes, int n_in,
                              void* d_out, int out_size, void* d_ws, size_t ws_size,
                              hipStream_t stream) {
  (void)in_sizes; (void)n_in; (void)out_size; (void)d_ws; (void)ws_size;
  const float* x = (const float*)d_in[0];
  float* out = (float*)d_out;

  dim3 grid(W / TILE_W, H / TILE_H, 16 * 3);  // 4 x 32 x 48 = 6144 blocks
  dim3 block(256);
  median3x3_kernel<<<grid, block, 0, stream>>>(x, out);
}